// SparseLinear_88064009437933
// MI455X (gfx1250) — compile-verified
//
#include <hip/hip_runtime.h>

// ---------------------------------------------------------------------------
// Dynamic-K sparse linear for MI455X (gfx1250, wave32, WMMA).
//   y[t*64+b, o] = sum_{c in active(t)} x[t*64+b, c] * W[o, c] + bias[o]
// active(t) = { c : max_b |x[t*64+b, c]| > 1e-6 }
//
// Kernel 1: per-tile prescan -> ordered compacted active-channel index list.
// Kernel 2: WMMA f32 16x16x4 GEMM over the compacted K.
//   - LDS double-buffered; full chunks staged with per-lane
//     global_load_async_to_lds_b32 (ASYNCcnt), overlapped with compute.
//   - K stored pair-interleaved in LDS so every WMMA fragment is one aligned
//     ds_load_b64 straight into an even-aligned VGPR pair (no repack movs).
// ---------------------------------------------------------------------------

typedef __attribute__((ext_vector_type(2))) float v2f;
typedef __attribute__((ext_vector_type(8))) float v8f;

#define THRESH   1e-6f
#define BM       64          // rows per tile (prescan granularity)
#define BN       128         // output cols per block
#define KC       32          // active channels staged per chunk (8 WMMA steps)
#define MAX_CIN  4096
#define PA       80          // BM padded: 2*PA ≡ 32 (mod 64) banks -> halves disjoint
#define PB       144         // BN padded: 2*PB ≡ 32 (mod 64)

// ---------------------------------------------------------------------------
// Kernel 1: prescan. One block (256 threads) per 64-row tile.
// ---------------------------------------------------------------------------
__global__ __launch_bounds__(256)
void SparseLinear_prescan(const float* __restrict__ x,
                          int* __restrict__ counts,
                          int* __restrict__ idxlist,
                          int c_in) {
    __shared__ unsigned char sFlag[MAX_CIN];
    __shared__ int sPart[256];

    const int tile = blockIdx.x;
    const int tid  = threadIdx.x;
    const long long row0 = (long long)tile * BM;
    const int cpt = c_in >> 8;                  // channels per thread (16)

    // Pass 1: strided channel assignment -> coalesced loads across the block.
    for (int j = 0; j < cpt; ++j) {
        const int c = j * 256 + tid;
        float m = 0.f;
        #pragma unroll 4
        for (int r = 0; r < BM; ++r) {
            m = fmaxf(m, fabsf(x[(row0 + r) * c_in + c]));
        }
        sFlag[c] = (m > THRESH) ? 1 : 0;
    }
    __syncthreads();

    // Pass 2: contiguous chunk per thread, ordered (deterministic) compaction.
    const int base = tid * cpt;
    int cnt = 0;
    #pragma unroll
    for (int j = 0; j < 16; ++j) cnt += sFlag[base + j];
    sPart[tid] = cnt;
    __syncthreads();

    for (int off = 1; off < 256; off <<= 1) {   // inclusive block scan
        const int v = sPart[tid];
        const int u = (tid >= off) ? sPart[tid - off] : 0;
        __syncthreads();
        sPart[tid] = v + u;
        __syncthreads();
    }

    int* dst = idxlist + (long long)tile * c_in;
    int p = sPart[tid] - cnt;                   // exclusive prefix
    #pragma unroll
    for (int j = 0; j < 16; ++j) {
        if (sFlag[base + j]) dst[p++] = base + j;
    }
    if (tid == 255) counts[tile] = sPart[255];
}

// ---------------------------------------------------------------------------
// Kernel 2: Dynamic-K GEMM. Grid (c_out/BN, n_tiles), 256 threads = 8 waves.
// Wave (wm, wn): wm in {0,16,32,48}, wn in {0,64}; 16x64 strip = 4 x v8f acc.
// LDS layout: K pair-interleaved, [pair][row/col][2], pair p = channels 2p,2p+1.
// ---------------------------------------------------------------------------
__global__ __launch_bounds__(256)
void SparseLinear_gemm_wmma(const float* __restrict__ x,
                            const float* __restrict__ w,
                            const float* __restrict__ bias,
                            float* __restrict__ y,
                            const int* __restrict__ counts,
                            const int* __restrict__ idxlist,
                            int c_in, int c_out) {
    __shared__ float sA[2][KC / 2][PA][2];   // 2*16*80*2*4  = 20.0 KB
    __shared__ float sB[2][KC / 2][PB][2];   // 2*16*144*2*4 = 36.0 KB

    const int tile = blockIdx.y;
    const int col0 = blockIdx.x * BN;
    const long long row0 = (long long)tile * BM;

    const int tid  = threadIdx.x;
    const int lane = tid & 31;
    const int wv   = tid >> 5;            // wave id 0..7
    const int wm   = (wv & 3) * 16;       // wave M offset
    const int wn   = (wv >> 2) * 64;      // wave N offset
    const int half = lane >> 4;           // 0: lanes 0-15, 1: lanes 16-31
    const int lr   = lane & 15;

    const int kcount  = counts[tile];
    const int* cidx   = idxlist + (long long)tile * c_in;
    const int nchunks = (kcount + KC - 1) / KC;
    const int full    = kcount / KC;      // chunks with no predication needed

    v8f acc[4];
    #pragma unroll
    for (int i = 0; i < 4; ++i)
        #pragma unroll
        for (int j = 0; j < 8; ++j) acc[i][j] = 0.f;

    // --- async gather of one full KC-chunk straight into LDS (per-lane DMA)
    auto stage_async = [&](int buf, int k0) {
        #pragma unroll
        for (int i = 0; i < (KC * BM) / 256; ++i) {   // A: 8 per thread
            const int e  = tid + i * 256;
            const int kk = e >> 6;
            const int r  = e & (BM - 1);
            const int ci = cidx[k0 + kk];
            const float* gp = &x[(row0 + r) * c_in + ci];
            const unsigned la =
                (unsigned)(unsigned long long)&sA[buf][kk >> 1][r][kk & 1];
            asm volatile("global_load_async_to_lds_b32 %0, %1, off"
                         :: "v"(la), "v"(gp) : "memory");
        }
        #pragma unroll
        for (int i = 0; i < (KC * BN) / 256; ++i) {   // B: 16 per thread
            const int e  = tid + i * 256;
            const int kk = e >> 7;
            const int n  = e & (BN - 1);
            const int ci = cidx[k0 + kk];
            const float* gp = &w[(long long)(col0 + n) * c_in + ci];
            const unsigned la =
                (unsigned)(unsigned long long)&sB[buf][kk >> 1][n][kk & 1];
            asm volatile("global_load_async_to_lds_b32 %0, %1, off"
                         :: "v"(la), "v"(gp) : "memory");
        }
    };

    // --- predicated VGPR-path gather for the (single) tail chunk, zero-fill
    auto stage_tail = [&](int buf, int k0) {
        #pragma unroll
        for (int i = 0; i < (KC * BM) / 256; ++i) {
            const int e  = tid + i * 256;
            const int kk = e >> 6;
            const int r  = e & (BM - 1);
            const int kg = k0 + kk;
            float v = 0.f;
            if (kg < kcount) v = x[(row0 + r) * c_in + cidx[kg]];
            sA[buf][kk >> 1][r][kk & 1] = v;
        }
        #pragma unroll
        for (int i = 0; i < (KC * BN) / 256; ++i) {
            const int e  = tid + i * 256;
            const int kk = e >> 7;
            const int n  = e & (BN - 1);
            const int kg = k0 + kk;
            float v = 0.f;
            if (kg < kcount) v = w[(long long)(col0 + n) * c_in + cidx[kg]];
            sB[buf][kk >> 1][n][kk & 1] = v;
        }
    };

    auto stage = [&](int buf, int c) {
        if (c < full) stage_async(buf, c * KC);
        else          stage_tail(buf, c * KC);
    };

    // Drain this wave's async DMAs, then block barrier.
    auto sync_stage = [&]() {
        asm volatile("s_wait_asynccnt 0x0" ::: "memory");
        __syncthreads();
    };

    // --- 8 WMMA K-steps over buffer `buf`; each fragment = one ds_load_b64.
    // V_WMMA_F32_16X16X4_F32 layout: lanes 0-15 hold K={4s,4s+1} (pair 2s),
    // lanes 16-31 hold K={4s+2,4s+3} (pair 2s+1).
    auto compute = [&](int buf) {
        #pragma unroll
        for (int step = 0; step < KC / 4; ++step) {
            const int p = step * 2 + half;
            const v2f a = *(const v2f*)&sA[buf][p][wm + lr][0];
            #pragma unroll
            for (int ns = 0; ns < 4; ++ns) {
                const v2f b = *(const v2f*)&sB[buf][p][wn + ns * 16 + lr][0];
                acc[ns] = __builtin_amdgcn_wmma_f32_16x16x4_f32(
                    /*neg_a=*/false, a, /*neg_b=*/false, b,
                    /*c_mod=*/(short)0, acc[ns],
                    /*reuse_a=*/false, /*reuse_b=*/false);
            }
        }
    };

    if (nchunks > 0) {
        stage(0, 0);
        sync_stage();
        int buf = 0;
        for (int c = 0; c < nchunks; ++c) {
            if (c + 1 < nchunks) stage(buf ^ 1, c + 1);  // DMA overlaps compute
            compute(buf);
            sync_stage();
            buf ^= 1;
        }
    }

    // Epilogue: bias add + store. D layout: VGPR i -> M = wm + half*8 + i,
    // N = lane%16.
    #pragma unroll
    for (int ns = 0; ns < 4; ++ns) {
        const int ncol = col0 + wn + ns * 16 + lr;
        const float bv = bias[ncol];
        #pragma unroll
        for (int i = 0; i < 8; ++i) {
            const int m = wm + half * 8 + i;
            y[(row0 + m) * c_out + ncol] = acc[ns][i] + bv;
        }
    }
}

// ---------------------------------------------------------------------------
extern "C" void kernel_launch(void* const* d_in, const int* in_sizes, int n_in,
                              void* d_out, int out_size, void* d_ws, size_t ws_size,
                              hipStream_t stream) {
    const float* x    = (const float*)d_in[0];
    const float* w    = (const float*)d_in[1];
    const float* bias = (const float*)d_in[2];
    float* y          = (float*)d_out;

    const int c_out   = in_sizes[2];                 // 4096
    const int c_in    = in_sizes[1] / c_out;         // 4096
    const int n       = in_sizes[0] / c_in;          // 16384
    const int n_tiles = n / BM;                      // 256

    int* counts  = (int*)d_ws;
    int* idxlist = counts + ((n_tiles + 63) & ~63);  // 256B-aligned

    SparseLinear_prescan<<<n_tiles, 256, 0, stream>>>(x, counts, idxlist, c_in);

    dim3 grid(c_out / BN, n_tiles);
    SparseLinear_gemm_wmma<<<grid, 256, 0, stream>>>(x, w, bias, y,
                                                     counts, idxlist, c_in, c_out);
}